// DataAssociationLoss_32152125177947
// MI455X (gfx1250) — compile-verified
//
#include <hip/hip_runtime.h>
#include <hip/hip_bf16.h>
#include <math.h>

#define BS 8
#define NX 2048
#define NY 2048
#define DD 256
#define EPSC 1e-8f
#define MARGINC 0.5f

#define SA_STRIDE 264   // floats; 1056 B row stride (16B aligned)
#define SB_STRIDE 72    // floats; 288 B row stride (16B aligned)
#define NCHUNK 128      // columns per LDS B chunk
#define KCHUNK 64       // K per LDS B chunk
#define NCHUNKS ((NY / NCHUNK) * (DD / KCHUNK))   // 64 total chunks

typedef __attribute__((ext_vector_type(2))) float v2f;
typedef __attribute__((ext_vector_type(8))) float v8f;

// Low 32 bits of a flat pointer into __shared__ are the LDS byte offset
// (flat->LDS mapping truncates to addr[31:0], ISA 10.2 aperture rules).
static __device__ __forceinline__ unsigned lds_off(const void* p) {
    return (unsigned)(uintptr_t)p;
}

// gfx1250 async copy: per-lane 16B global -> LDS, tracked by ASYNCcnt.
static __device__ __forceinline__ void async_ld_b128(unsigned lds, const float* g) {
    asm volatile("global_load_async_to_lds_b128 %0, %1, off"
                 :: "v"(lds), "v"(g) : "memory");
}
static __device__ __forceinline__ void wait_async0() {
    asm volatile("s_wait_asynccnt 0x0" ::: "memory");
}

// ---------------- row norms ----------------
__global__ __launch_bounds__(256) void da_norms(const float* __restrict__ first,
                                                const float* __restrict__ second,
                                                float* __restrict__ normX,
                                                float* __restrict__ normY) {
    int gw   = (blockIdx.x * 256 + threadIdx.x) >> 5;
    int lane = threadIdx.x & 31;
    const float* src;
    float* dst;
    if (gw < BS * NX) { src = first  + (size_t)gw * DD;  dst = normX + gw; }
    else              { int r = gw - BS * NX;
                        src = second + (size_t)r * DD;   dst = normY + r;  }
    float s = 0.f;
#pragma unroll
    for (int q = 0; q < DD / 32; ++q) { float v = src[lane + q * 32]; s += v * v; }
#pragma unroll
    for (int d = 1; d < 32; d <<= 1) s += __shfl_xor(s, d, 32);
    if (lane == 0) *dst = sqrtf(s);
}

// ---------------- target lookup (first match == argmax-of-bool) ----------------
__global__ __launch_bounds__(256) void da_target(const int* __restrict__ fids,
                                                 const int* __restrict__ sids,
                                                 int* __restrict__ tgt) {
    int id = blockIdx.x * 256 + threadIdx.x;
    if (id >= BS * NX) return;
    int b = id >> 11;
    int v = fids[id];
    const int* s = sids + (b << 11);
    int t = NY;
    for (int j = 0; j < NY; ++j) {
        if (s[j] == v) { t = j; break; }
    }
    tgt[id] = t;
}

// ---------------- fused GEMM + losses ----------------
__global__ __launch_bounds__(256) void da_gemm_loss(
    const float* __restrict__ first, const float* __restrict__ second,
    const float* __restrict__ missed,
    const float* __restrict__ normX, const float* __restrict__ normY,
    const int* __restrict__ target,
    float* __restrict__ out,            // d_out base: [0..2]=losses, [3..]=aff
    float* __restrict__ cosP, float* __restrict__ bceP, float* __restrict__ ceP) {

    __shared__ float sA[16 * SA_STRIDE];
    __shared__ float sB[2][NCHUNK * SB_STRIDE];
    __shared__ float sM[128], sS[128], sC[128], sE[128], sT[128];

    const int mblk = blockIdx.x;        // 0..127 (16-row block)
    const int b    = blockIdx.y;        // 0..7
    const int tid  = threadIdx.x;
    const int lane = tid & 31;
    const int wave = tid >> 5;          // 0..7
    const int hi   = lane >> 4;         // 0/1
    const int nlo  = lane & 15;
    float* aff = out + 3;

    // ---- issue async stage of A block: 16 rows x 256 K (4 x b128 per thread) ----
    {
        const float* F = first + ((size_t)b * NX + (size_t)mblk * 16) * DD;
        int row = tid >> 4;
        int seg = tid & 15;
        const float* g = F + (size_t)row * DD + seg * 16;
        unsigned l = lds_off(&sA[row * SA_STRIDE + seg * 16]);
#pragma unroll
        for (int q = 0; q < 4; ++q) async_ld_b128(l + q * 16, g + q * 4);
    }
    // ---- issue async stage of B chunk 0 (8 x b128 per thread) ----
    const int colL = tid >> 1;
    const int half = tid & 1;
    {
        const float* S = second + ((size_t)b * NY) * DD;   // nc=0, kc=0
        const float* g = S + (size_t)colL * DD + half * 32;
        unsigned l = lds_off(&sB[0][colL * SB_STRIDE + half * 32]);
#pragma unroll
        for (int q = 0; q < 8; ++q) async_ld_b128(l + q * 16, g + q * 4);
    }

    // per-lane row constants (rows hi*8 .. hi*8+7 of this block)
    float nxr[8];
    int   tg[8];
#pragma unroll
    for (int r = 0; r < 8; ++r) {
        int rowG = mblk * 16 + hi * 8 + r;
        nxr[r] = normX[b * NX + rowG];
        tg[r]  = target[b * NX + rowG];
    }

    float mrow[8], srow[8], crow[8], erow[8], trow[8];
#pragma unroll
    for (int r = 0; r < 8; ++r) {
        mrow[r] = -INFINITY; srow[r] = 0.f; crow[r] = 0.f; erow[r] = 0.f; trow[r] = -INFINITY;
    }

    // ---- prologue: A block + B chunk 0 landed in LDS ----
    wait_async0();
    __syncthreads();

    // ---- hoist A fragments into registers: areg[kc*16+ks], constant-indexed ----
    v2f areg[64];
    {
        const float* aRow = &sA[nlo * SA_STRIDE + 2 * hi];
#pragma unroll
        for (int kc = 0; kc < 4; ++kc)
#pragma unroll
            for (int ks = 0; ks < 16; ++ks)
                areg[kc * 16 + ks] = *(const v2f*)(aRow + kc * KCHUNK + ks * 4);
    }

    const v8f vzero = {};

    for (int nc = 0; nc < NY / NCHUNK; ++nc) {
        v8f acc = vzero;
#pragma unroll
        for (int kc = 0; kc < 4; ++kc) {
            // chunk c = nc*4 + kc lives in buffer (kc & 1); its loads were issued
            // one stage earlier (or in the prologue for c == 0).
            wait_async0();      // my loads for this chunk are in LDS
            __syncthreads();    // everyone's landed; prefetch target buffer is free

            {   // prefetch chunk c+1 into the other buffer
                const int c1 = nc * 4 + kc + 1;
                if (c1 < NCHUNKS) {
                    const int nc1 = c1 >> 2;
                    const int kc1 = c1 & 3;
                    const float* S = second + ((size_t)b * NY + (size_t)nc1 * NCHUNK) * DD
                                   + kc1 * KCHUNK;
                    const float* g = S + (size_t)colL * DD + half * 32;
                    unsigned l = lds_off(&sB[(kc + 1) & 1][colL * SB_STRIDE + half * 32]);
#pragma unroll
                    for (int q = 0; q < 8; ++q) async_ld_b128(l + q * 16, g + q * 4);
                }
            }

            // ---- 16 chained WMMAs; A from registers, B from LDS ----
            const float* bCol = &sB[kc & 1][(wave * 16 + nlo) * SB_STRIDE + 2 * hi];
#pragma unroll
            for (int ks = 0; ks < 16; ++ks) {
                v2f bv = *(const v2f*)(bCol + ks * 4);
                acc = __builtin_amdgcn_wmma_f32_16x16x4_f32(
                    false, areg[kc * 16 + ks], false, bv, (short)0, acc, false, false);
            }
        }

        // ---- epilogue for this wave's 16x16 tile (full K accumulated) ----
        {
            int colG = nc * NCHUNK + wave * 16 + nlo;
            float nyc = normY[b * NY + colG];
#pragma unroll
            for (int r = 0; r < 8; ++r) {
                int rowG = mblk * 16 + hi * 8 + r;
                float val = acc[r] / fmaxf(nxr[r] * nyc, EPSC);
                aff[((size_t)(b * NX + rowG)) * (NY + 1) + colG] = val;
                bool pos = (tg[r] == colG);
                crow[r] += pos ? (1.f - val) : fmaxf(val - MARGINC, 0.f);
                float y = pos ? 1.f : 0.f;
                erow[r] += fmaxf(val, 0.f) - val * y + log1pf(expf(-fabsf(val)));
                float mN = fmaxf(mrow[r], val);
                srow[r] = srow[r] * expf(mrow[r] - mN) + expf(val - mN);
                mrow[r] = mN;
                if (pos) trow[r] = val;
            }
        }
    }

    // cross-lane reduction within each 16-lane group
#pragma unroll
    for (int r = 0; r < 8; ++r) {
#pragma unroll
        for (int d = 1; d < 16; d <<= 1) {
            float mo = __shfl_xor(mrow[r], d, 32);
            float so = __shfl_xor(srow[r], d, 32);
            float mN = fmaxf(mrow[r], mo);
            srow[r] = srow[r] * expf(mrow[r] - mN) + so * expf(mo - mN);
            mrow[r] = mN;
            crow[r] += __shfl_xor(crow[r], d, 32);
            erow[r] += __shfl_xor(erow[r], d, 32);
            trow[r] = fmaxf(trow[r], __shfl_xor(trow[r], d, 32));
        }
    }
    if (nlo == 0) {
        int base = wave * 16 + hi * 8;
#pragma unroll
        for (int r = 0; r < 8; ++r) {
            sM[base + r] = mrow[r]; sS[base + r] = srow[r];
            sC[base + r] = crow[r]; sE[base + r] = erow[r]; sT[base + r] = trow[r];
        }
    }
    __syncthreads();

    // cross-wave combine: threads 0..15 each own one row
    if (tid < 16) {
        int row = tid;
        float m = -INFINITY, s = 0.f, c2 = 0.f, e = 0.f, at = -INFINITY;
#pragma unroll
        for (int w = 0; w < 8; ++w) {
            float mo = sM[w * 16 + row], so = sS[w * 16 + row];
            float mN = fmaxf(m, mo);
            s = s * expf(m - mN) + so * expf(mo - mN);
            m = mN;
            c2 += sC[w * 16 + row];
            e  += sE[w * 16 + row];
            at = fmaxf(at, sT[w * 16 + row]);
        }
        // fold in the missed-detection (delta) column
        float delta = missed[0];
        int rowG = mblk * 16 + row;
        int t = target[b * NX + rowG];
        float mN = fmaxf(m, delta);
        s = s * expf(m - mN) + expf(delta - mN);
        m = mN;
        float yD = (t == NY) ? 1.f : 0.f;
        e += fmaxf(delta, 0.f) - delta * yD + log1pf(expf(-fabsf(delta)));
        if (t == NY) at = delta;
        float ce = (m + logf(s)) - at;               // -logp[target]
        aff[((size_t)(b * NX + rowG)) * (NY + 1) + NY] = delta;
#pragma unroll
        for (int d = 1; d < 16; d <<= 1) {
            c2 += __shfl_xor(c2, d, 32);
            e  += __shfl_xor(e,  d, 32);
            ce += __shfl_xor(ce, d, 32);
        }
        if (tid == 0) {
            int wg = b * gridDim.x + mblk;
            cosP[wg] = c2; bceP[wg] = e; ceP[wg] = ce;
        }
    }
}

// ---------------- final weighted reduction ----------------
__global__ __launch_bounds__(32) void da_final(const float* __restrict__ cosP,
                                               const float* __restrict__ bceP,
                                               const float* __restrict__ ceP,
                                               float* __restrict__ out) {
    int b = threadIdx.x;
    float c = 0.f, e = 0.f, x = 0.f;
    if (b < BS) {
        for (int k = 0; k < NX / 16; ++k) {
            int i = b * (NX / 16) + k;
            c += cosP[i]; e += bceP[i]; x += ceP[i];
        }
        float w = powf((float)BS, (float)b - (float)BS);   // 8^(b-8)
        c = w * (c / (float)NY);
        e = w * (e / ((float)NX * (float)(NY + 1)));
        x = w * (x / (float)NX);
    }
#pragma unroll
    for (int d = 1; d < 8; d <<= 1) {
        c += __shfl_xor(c, d, 32);
        e += __shfl_xor(e, d, 32);
        x += __shfl_xor(x, d, 32);
    }
    if (b == 0) { out[0] = c; out[1] = e; out[2] = x; }
}

extern "C" void kernel_launch(void* const* d_in, const int* in_sizes, int n_in,
                              void* d_out, int out_size, void* d_ws, size_t ws_size,
                              hipStream_t stream) {
    (void)in_sizes; (void)n_in; (void)out_size; (void)ws_size;
    const float* first  = (const float*)d_in[0];
    const int*   fids   = (const int*)d_in[1];
    const float* second = (const float*)d_in[2];
    const int*   sids   = (const int*)d_in[3];
    const float* missed = (const float*)d_in[4];
    float* out = (float*)d_out;

    float* wsf   = (float*)d_ws;
    float* normX = wsf;                         // BS*NX
    float* normY = wsf + BS * NX;               // BS*NY
    int*   tgt   = (int*)(wsf + 2 * BS * NX);   // BS*NX ints
    float* cosP  = wsf + 3 * BS * NX;           // BS*(NX/16)
    float* bceP  = cosP + BS * (NX / 16);
    float* ceP   = bceP + BS * (NX / 16);

    da_norms<<<(2 * BS * NX) / 8, 256, 0, stream>>>(first, second, normX, normY);
    da_target<<<(BS * NX) / 256, 256, 0, stream>>>(fids, sids, tgt);
    da_gemm_loss<<<dim3(NX / 16, BS), 256, 0, stream>>>(first, second, missed,
                                                        normX, normY, tgt,
                                                        out, cosP, bceP, ceP);
    da_final<<<1, 32, 0, stream>>>(cosP, bceP, ceP, out);
}